// FractalDimensionLayer_52235392254132
// MI455X (gfx1250) — compile-verified
//
#include <hip/hip_runtime.h>
#include <math.h>

#define MAX_K   10
#define BUF     50
#define HIDDEN  512
#define SEQ     4096
#define P_TOTAL (16 * 4096)

typedef __attribute__((ext_vector_type(2))) float v2f;
typedef __attribute__((ext_vector_type(4))) float v4f;
typedef __attribute__((ext_vector_type(8))) float v8f;

// -------------------------------------------------------------------------
// Phase 1: per-position Higuchi fractal dimension -> masked h[p] in d_ws.
// One thread per position p. Block stages close[] window slab into LDS.
// -------------------------------------------------------------------------
__global__ __launch_bounds__(256) void hfd_kernel(const float* __restrict__ x,
                                                  float* __restrict__ hout) {
    __shared__ float sm[256 + BUF - 1]; // close[base-49 .. base+255]
    const int tid  = threadIdx.x;
    const int base = blockIdx.x * 256;

    // Stage close = x[:, :, 3] (stride-8 floats), zero-pad before global start.
    {
        int g = base - (BUF - 1) + tid;
        sm[tid] = (g >= 0) ? x[g * 8 + 3] : 0.0f;
        if (tid < BUF - 1) {
            int g2 = base - (BUF - 1) + 256 + tid;   // always >= 0, <= P-1
            sm[256 + tid] = x[g2 * 8 + 3];
        }
    }
    __syncthreads();

    // Copy this thread's 50-sample window to registers.
    float w[BUF];
#pragma unroll
    for (int j = 0; j < BUF; ++j) w[j] = sm[tid + j];

    // Higuchi curve lengths L(k), k = 1..10 (fully unrolled, constant bounds).
    float lk[MAX_K];
#pragma unroll
    for (int k = 1; k <= MAX_K; ++k) {
        float acc = 0.0f;
#pragma unroll
        for (int m = 0; m < k; ++m) {
            const int len = (BUF - 1 - m) / k + 1;  // len(np.arange(m,50,k))
            if (len >= 2) {
                float prev = w[m];
                float lengths = 0.0f;
#pragma unroll
                for (int i = 1; i < len; ++i) {
                    float cur = w[m + i * k];
                    lengths += fabsf(cur - prev);
                    prev = cur;
                }
                acc += lengths * ((float)(BUF - 1) / (float)(len * k));
            }
        }
        lk[k - 1] = acc / (float)k;
    }

    // Weighted log-log regression; slope -> -HFD.
    const float LOGK[MAX_K] = {
        0.0f,          0.69314718f, 1.09861229f, 1.38629436f, 1.60943791f,
        1.79175947f,   1.94591015f, 2.07944154f, 2.19722458f, 2.30258509f };

    float Sw = 0.f, Sx = 0.f, Sy = 0.f, Sxx = 0.f, Sxy = 0.f;
#pragma unroll
    for (int k = 0; k < MAX_K; ++k) {
        float wgt   = (lk[k] > 0.0f) ? 1.0f : 0.0f;
        float logk  = LOGK[k];
        float loglk = logf(fmaxf(lk[k], 1e-30f));
        loglk = (lk[k] > 0.0f) ? loglk : 0.0f;
        Sw  += wgt;
        Sx  += wgt * logk;
        Sy  += wgt * loglk;
        Sxx += wgt * logk * logk;
        Sxy += wgt * logk * loglk;
    }
    float denom = Sw * Sxx - Sx * Sx;
    float safe  = fabsf(denom) > 1e-12f ? denom : 1.0f;
    float slope = (fabsf(denom) > 1e-12f) ? (Sw * Sxy - Sx * Sy) / safe : 0.0f;
    float hfd   = (Sw > 1.0f) ? -slope : 0.0f;

    const int p = base + tid;
    float mask  = ((p & (SEQ - 1)) >= MAX_K) ? 1.0f : 0.0f;
    hout[p] = hfd * mask;
}

// -------------------------------------------------------------------------
// Phase 2 (transposed tiles): D[m][n] = W[c0+m]*h[p0+n] + b[c0+m], relu, store.
//   A (16x4): column K=0 holds W[c0..c0+15]  -> changes per hidden tile
//   B (4x16): row    K=0 holds h[p0..p0+15]  -> loop-invariant per wave
//   C: bias, 8 consecutive channels per lane-half -> two ds_load_b128
// D layout: VGPR v, lane L -> (M = v + (L<16?0:8), N = L&15), so each lane
// owns out[p0+(L&15)][c0+mofs .. +7] = 8 CONSECUTIVE floats -> 2x b128 store.
// Store floor: 128 MiB -> ~5.6 us @ 23.3 TB/s; this keeps the store path at
// 2 wide stores per WMMA tile instead of 8 narrow ones.
// -------------------------------------------------------------------------
__global__ __launch_bounds__(256) void outer_wmma_kernel(const float* __restrict__ h,
                                                         const float* __restrict__ W,
                                                         const float* __restrict__ b,
                                                         float* __restrict__ out) {
    __shared__ float sW[HIDDEN];
    __shared__ float sB[HIDDEN];
    const int tid = threadIdx.x;
    sW[tid]       = W[tid];
    sW[tid + 256] = W[tid + 256];
    sB[tid]       = b[tid];
    sB[tid + 256] = b[tid + 256];
    __syncthreads();

    const int lane  = tid & 31;
    const int wave  = tid >> 5;
    const int ptile = blockIdx.x * 8 + wave;     // 0..4095 position tiles
    const int p0    = ptile * 16;
    const int nidx  = lane & 15;
    const bool lo   = (lane < 16);
    const int mofs  = lo ? 0 : 8;

    // B fragment (h values), loop-invariant: lanes 0-15 VGPR0 = K=0 row.
    float hv = h[p0 + nidx];
    v2f Bf;
    Bf.x = lo ? hv : 0.0f;
    Bf.y = 0.0f;

    // Each lane's output row base: out[p0 + nidx][...]
    float* rowbase = out + (size_t)(p0 + nidx) * HIDDEN;

    for (int ct = 0; ct < HIDDEN / 16; ++ct) {
        const int c0 = ct * 16;

        // A fragment (W values): lanes 0-15 hold M=lane, K=0 column.
        float wv = sW[c0 + (lane & 15)];
        v2f Af;
        Af.x = lo ? wv : 0.0f;
        Af.y = 0.0f;

        // C = bias: lane-half owns channels c0+mofs .. c0+mofs+7 (32B aligned).
        const float4 blo = *(const float4*)&sB[c0 + mofs];
        const float4 bhi = *(const float4*)&sB[c0 + mofs + 4];
        v8f C = { blo.x, blo.y, blo.z, blo.w, bhi.x, bhi.y, bhi.z, bhi.w };

        v8f D = __builtin_amdgcn_wmma_f32_16x16x4_f32(
            /*neg_a=*/false, Af, /*neg_b=*/false, Bf,
            /*c_mod=*/(short)0, C, /*reuse_a=*/false, /*reuse_b=*/false);

        // relu (single select per element) and two 128-bit stores per lane.
        float4 r0, r1;
        r0.x = D[0] > 0.0f ? D[0] : 0.0f;
        r0.y = D[1] > 0.0f ? D[1] : 0.0f;
        r0.z = D[2] > 0.0f ? D[2] : 0.0f;
        r0.w = D[3] > 0.0f ? D[3] : 0.0f;
        r1.x = D[4] > 0.0f ? D[4] : 0.0f;
        r1.y = D[5] > 0.0f ? D[5] : 0.0f;
        r1.z = D[6] > 0.0f ? D[6] : 0.0f;
        r1.w = D[7] > 0.0f ? D[7] : 0.0f;

        *(float4*)&rowbase[c0 + mofs]     = r0;
        *(float4*)&rowbase[c0 + mofs + 4] = r1;
    }
}

// -------------------------------------------------------------------------
extern "C" void kernel_launch(void* const* d_in, const int* in_sizes, int n_in,
                              void* d_out, int out_size, void* d_ws, size_t ws_size,
                              hipStream_t stream) {
    const float* x = (const float*)d_in[0];   // (16, 4096, 8)
    const float* W = (const float*)d_in[1];   // (512, 1)
    const float* b = (const float*)d_in[2];   // (512,)
    float* out  = (float*)d_out;              // (16, 4096, 512)
    float* hbuf = (float*)d_ws;               // 65536 floats = 256 KB scratch

    hipLaunchKernelGGL(hfd_kernel,
                       dim3(P_TOTAL / 256), dim3(256), 0, stream, x, hbuf);

    // 4096 position tiles, 8 waves per 256-thread block -> 512 blocks.
    hipLaunchKernelGGL(outer_wmma_kernel,
                       dim3(P_TOTAL / 16 / 8), dim3(256), 0, stream,
                       hbuf, W, b, out);
}